// GAT_49271864819841
// MI455X (gfx1250) — compile-verified
//
#include <hip/hip_runtime.h>
#include <hip/hip_bf16.h>
#include <math.h>

typedef float v2f __attribute__((ext_vector_type(2)));
typedef float v8f __attribute__((ext_vector_type(8)));

#define FDIM 128
#define ATT 64
#define UNITSX 128
#define HEADS 8
#define QKV_LD 256   // Q[0:64) K[64:128) V[128:256)
#define XS_LD 132    // padded LDS stride (bank-conflict free)

// ---------------- init ----------------
__global__ void gat_init(float* __restrict__ out, float* __restrict__ smax,
                         float* __restrict__ ssum, int N) {
  int i = blockIdx.x * blockDim.x + threadIdx.x;
  if (i < N * 128) out[i] = 0.0f;
  if (i < N * 8) { smax[i] = -__builtin_inff(); ssum[i] = 0.0f; }
}

// ---------------- fused QKV GEMM (fp32 WMMA 16x16x4) ----------------
// One block = 16 node rows. 4 waves; each wave owns 4 of the 16 col-tiles
// of the fused [128 x 256] weight (Q:0-3, K:4-7, V:8-15).
__global__ __launch_bounds__(128)
void gat_qkv_wmma(const float* __restrict__ x,
                  const float* __restrict__ wq, const float* __restrict__ bq,
                  const float* __restrict__ wk, const float* __restrict__ bk,
                  const float* __restrict__ wv,
                  float* __restrict__ qkv, int N) {
  __shared__ float xs[16 * XS_LD];
  const int row0 = blockIdx.x * 16;
  const int t = threadIdx.x;

  // stage 16x128 X tile into LDS (row clamped for safety; N%16==0 normally)
  {
    int m = t >> 3;                // 0..15
    int cc = (t & 7) * 16;         // col chunk of 16 floats
    int r = row0 + m; if (r >= N) r = N - 1;
    const float4* src = (const float4*)(x + (size_t)r * FDIM + cc);
    float4 v0 = src[0], v1 = src[1], v2 = src[2], v3 = src[3];
    float4* dst = (float4*)(xs + m * XS_LD + cc);
    dst[0] = v0; dst[1] = v1; dst[2] = v2; dst[3] = v3;
  }
  __syncthreads();

  const int wave = t >> 5;
  const int lane = t & 31;
  const int n    = lane & 15;   // B/D column within tile
  const int kh   = lane >> 4;   // lane half -> K pair / M offset

  const float* arow = xs + (lane & 15) * XS_LD + 2 * kh;  // A: m = lane&15

  for (int cti = 0; cti < 4; ++cti) {
    const int ct = wave * 4 + cti;           // uniform within wave
    const float* W; int ldw, cb; bool dorelu; float cinit;
    if (ct < 4)      { W = wq; ldw = ATT;    cb = ct * 16;       dorelu = true;  cinit = bq[cb + n]; }
    else if (ct < 8) { W = wk; ldw = ATT;    cb = (ct - 4) * 16; dorelu = true;  cinit = bk[cb + n]; }
    else             { W = wv; ldw = UNITSX; cb = (ct - 8) * 16; dorelu = false; cinit = 0.0f; }

    v8f c;
#pragma unroll
    for (int v = 0; v < 8; ++v) c[v] = cinit;   // C(m,n) = bias[n], all m

    const float* bcol = W + (size_t)(2 * kh) * ldw + cb + n;

#pragma unroll
    for (int k0 = 0; k0 < FDIM; k0 += 4) {
      v2f a, b;
      a[0] = arow[k0 + 0];                 // A(m, k0+2kh+0)
      a[1] = arow[k0 + 1];                 // A(m, k0+2kh+1)
      b[0] = bcol[(size_t)k0 * ldw];       // B(k0+2kh+0, n)
      b[1] = bcol[(size_t)(k0 + 1) * ldw]; // B(k0+2kh+1, n)
      c = __builtin_amdgcn_wmma_f32_16x16x4_f32(
              /*neg_a=*/false, a, /*neg_b=*/false, b,
              /*c_mod=*/(short)0, c, /*reuse_a=*/false, /*reuse_b=*/false);
    }

    // D layout: VGPR v -> rows v (lanes 0-15) and v+8 (lanes 16-31)
#pragma unroll
    for (int v = 0; v < 8; ++v) {
      int m = v + 8 * kh;
      int r = row0 + m;
      if (r < N) {
        float val = c[v];
        if (dorelu) val = fmaxf(val, 0.0f);
        qkv[(size_t)r * QKV_LD + ct * 16 + n] = val;
      }
    }
  }
}

// float atomic max via signed/unsigned int ordering trick
__device__ __forceinline__ void atomicMaxF(float* addr, float v) {
  if (v >= 0.0f) atomicMax((int*)addr, __float_as_int(v));
  else           atomicMin((unsigned int*)addr, __float_as_uint(v));
}

// ---------------- per-edge score + segment max ----------------
__global__ void gat_edge_score(const float* __restrict__ qkv,
                               const int* __restrict__ ei,
                               int E, int N,
                               float* __restrict__ score,
                               float* __restrict__ smax) {
  int e = blockIdx.x * blockDim.x + threadIdx.x;
  int Etot = E + N;
  if (e >= Etot) return;
  int r, cidx;
  if (e < E) { r = ei[e]; cidx = ei[E + e]; } else { r = cidx = e - E; }

  const float4* Qp = (const float4*)(qkv + (size_t)r * QKV_LD);
  const float4* Kp = (const float4*)(qkv + (size_t)cidx * QKV_LD + 64);

#pragma unroll
  for (int h = 0; h < HEADS; ++h) {
    float4 q0 = Qp[h * 2], q1 = Qp[h * 2 + 1];
    float4 k0 = Kp[h * 2], k1 = Kp[h * 2 + 1];
    float s = q0.x * k0.x + q0.y * k0.y + q0.z * k0.z + q0.w * k0.w
            + q1.x * k1.x + q1.y * k1.y + q1.z * k1.z + q1.w * k1.w;
    score[(size_t)e * 8 + h] = s;
    atomicMaxF(&smax[(size_t)r * 8 + h], s);
  }
}

// ---------------- per-edge exp + unnormalized aggregation ----------------
__global__ void gat_edge_accum(const float* __restrict__ qkv,
                               const int* __restrict__ ei,
                               int E, int N,
                               const float* __restrict__ score,
                               const float* __restrict__ smax,
                               float* __restrict__ ssum,
                               float* __restrict__ out) {
  int e = blockIdx.x * blockDim.x + threadIdx.x;
  int Etot = E + N;
  if (e >= Etot) return;
  int r, cidx;
  if (e < E) { r = ei[e]; cidx = ei[E + e]; } else { r = cidx = e - E; }

  const float4* Vp = (const float4*)(qkv + (size_t)cidx * QKV_LD + 128);
  float* outp = out + (size_t)r * 128;

#pragma unroll
  for (int h = 0; h < HEADS; ++h) {
    float ex = __expf(score[(size_t)e * 8 + h] - smax[(size_t)r * 8 + h]);
    atomicAdd(&ssum[(size_t)r * 8 + h], ex);
#pragma unroll
    for (int j = 0; j < 4; ++j) {
      float4 v = Vp[h * 4 + j];
      atomicAdd(&outp[h * 16 + j * 4 + 0], ex * v.x);
      atomicAdd(&outp[h * 16 + j * 4 + 1], ex * v.y);
      atomicAdd(&outp[h * 16 + j * 4 + 2], ex * v.z);
      atomicAdd(&outp[h * 16 + j * 4 + 3], ex * v.w);
    }
  }
}

// ---------------- finalize: divide by segment sum, add bias ----------------
__global__ void gat_finalize(float* __restrict__ out,
                             const float* __restrict__ ssum,
                             const float* __restrict__ bias, int N) {
  int i = blockIdx.x * blockDim.x + threadIdx.x;
  if (i >= N * 128) return;
  int node = i >> 7;
  int c = i & 127;
  int h = c >> 4;
  out[i] = out[i] / ssum[(size_t)node * 8 + h] + bias[c];
}

extern "C" void kernel_launch(void* const* d_in, const int* in_sizes, int n_in,
                              void* d_out, int out_size, void* d_ws, size_t ws_size,
                              hipStream_t stream) {
  const float* x    = (const float*)d_in[0];
  const int*   ei   = (const int*)d_in[1];
  const float* wq   = (const float*)d_in[2];
  const float* bq   = (const float*)d_in[3];
  const float* wk   = (const float*)d_in[4];
  const float* bk   = (const float*)d_in[5];
  const float* wv   = (const float*)d_in[6];
  const float* bias = (const float*)d_in[7];

  const int N = in_sizes[0] / FDIM;
  const int E = in_sizes[1] / 2;
  const int Etot = E + N;

  float* ws    = (float*)d_ws;
  float* qkv   = ws;                         // N*256
  float* score = qkv + (size_t)N * QKV_LD;   // Etot*8
  float* smax  = score + (size_t)Etot * 8;   // N*8
  float* ssum  = smax + (size_t)N * 8;       // N*8
  float* out   = (float*)d_out;              // N*128 (also the accumulator)

  const int initN = N * 128;
  gat_init<<<(initN + 255) / 256, 256, 0, stream>>>(out, smax, ssum, N);
  gat_qkv_wmma<<<(N + 15) / 16, 128, 0, stream>>>(x, wq, bq, wk, bk, wv, qkv, N);
  gat_edge_score<<<(Etot + 255) / 256, 256, 0, stream>>>(qkv, ei, E, N, score, smax);
  gat_edge_accum<<<(Etot + 255) / 256, 256, 0, stream>>>(qkv, ei, E, N, score, smax, ssum, out);
  gat_finalize<<<(initN + 255) / 256, 256, 0, stream>>>(out, ssum, bias, N);
}